// GraphSAGE_29978871726568
// MI455X (gfx1250) — compile-verified
//
#include <hip/hip_runtime.h>

typedef __attribute__((ext_vector_type(2))) float v2f;
typedef __attribute__((ext_vector_type(4))) float v4f;
typedef __attribute__((ext_vector_type(8))) float v8f;

#define EPS 1e-12f

// ---------------------------------------------------------------------------
// Kernel 1: per-row inverse L2 norm. One wave32 per row.
// ---------------------------------------------------------------------------
__global__ void sage_rownorm(const float* __restrict__ x, float* __restrict__ invn,
                             int nrows, int f) {
    int row  = blockIdx.x * (blockDim.x >> 5) + (threadIdx.x >> 5);
    int lane = threadIdx.x & 31;
    if (row >= nrows) return;                     // wave-uniform
    const float* xr = x + (size_t)row * f;
    float s = 0.f;
    for (int k = lane; k < f; k += 32) { float v = xr[k]; s += v * v; }
    #pragma unroll
    for (int m = 16; m >= 1; m >>= 1) s += __shfl_xor(s, m);
    if (lane == 0) invn[row] = 1.0f / fmaxf(sqrtf(s), EPS);
}

// ---------------------------------------------------------------------------
// Kernel 2: pool GEMM.  H[n,o] = relu( (X[n,:]*invn[n]) . W[o,:] + b[o] )
//           HSQ = H*H   (MU = 2 fused here)
// One wave per 16x16 output tile, K stepped by 4 via V_WMMA_F32_16X16X4_F32.
// nrows % 16 == 0, K % 4 == 0, outc % 16 == 0 (holds: 50000/128/64).
// ---------------------------------------------------------------------------
__global__ void sage_pool_gemm(const float* __restrict__ X, const float* __restrict__ invn,
                               const float* __restrict__ W, const float* __restrict__ bias,
                               float* __restrict__ H, float* __restrict__ HSQ,
                               int nrows, int K, int outc) {
    const int lane = threadIdx.x & 31;
    const int wave = threadIdx.x >> 5;
    const int colTiles = outc >> 4;
    const int rowTiles = nrows >> 4;
    int tile = blockIdx.x * (blockDim.x >> 5) + wave;
    if (tile >= rowTiles * colTiles) return;      // wave-uniform
    const int rt = tile / colTiles, ct = tile % colTiles;
    const int row0 = rt << 4, c0 = ct << 4;
    const int half = lane >> 4;                   // selects K pair / row half
    const int lid  = lane & 15;

    // A: lane holds X[row0+lid][kk + 2*half + {0,1}] * invn
    const float* arow = X + (size_t)(row0 + lid) * K + 2 * half;
    const float  sc   = invn[row0 + lid];
    // B[k][n] = W[n][k], n = c0+lid
    const float* brow = W + (size_t)(c0 + lid) * K + 2 * half;

    v8f acc = {};
    for (int k = 0; k < K; k += 4) {
        v2f a, b;
        a.x = arow[k]     * sc;
        a.y = arow[k + 1] * sc;
        b.x = brow[k];
        b.y = brow[k + 1];
        acc = __builtin_amdgcn_wmma_f32_16x16x4_f32(false, a, false, b,
                                                    (short)0, acc, false, false);
    }

    const float bv = bias[c0 + lid];
    // D: lane l, vgpr j -> row = row0 + j + 8*half, col = c0 + lid
    float* hp = H   + (size_t)(row0 + 8 * half) * outc + c0 + lid;
    float* hq = HSQ + (size_t)(row0 + 8 * half) * outc + c0 + lid;
    #pragma unroll
    for (int j = 0; j < 8; ++j) {
        float v = fmaxf(acc[j] + bv, 0.f);
        hp[(size_t)j * outc] = v;
        hq[(size_t)j * outc] = v * v;
    }
}

// ---------------------------------------------------------------------------
// Kernel 3: zero the aggregation buffer.
// ---------------------------------------------------------------------------
__global__ void sage_zero(float* __restrict__ p, size_t n) {
    size_t i = (size_t)blockIdx.x * blockDim.x + threadIdx.x;
    size_t st = (size_t)gridDim.x * blockDim.x;
    for (; i < n; i += st) p[i] = 0.f;
}

// ---------------------------------------------------------------------------
// Kernel 4: SpMM scatter-add. One wave per edge; lane l handles 4 features
// via a float4 gather + 4 f32 atomics (L2-resident destination).
// F must be 128 (32 lanes * 4).
// ---------------------------------------------------------------------------
__global__ void sage_spmm(const float* __restrict__ HSQ, const int* __restrict__ src,
                          const int* __restrict__ dst, const float* __restrict__ val,
                          float* __restrict__ AGG, int E, int F) {
    int e    = (int)(((size_t)blockIdx.x * blockDim.x + threadIdx.x) >> 5);
    int lane = threadIdx.x & 31;
    if (e >= E) return;                           // wave-uniform
    const int   s = src[e];
    const int   d = dst[e];
    const float v = val[e];
    const v4f* sp = (const v4f*)(HSQ + (size_t)s * F);
    float*     dp = AGG + (size_t)d * F + lane * 4;
    v4f hv = sp[lane];
    atomicAdd(dp + 0, hv.x * v);
    atomicAdd(dp + 1, hv.y * v);
    atomicAdd(dp + 2, hv.z * v);
    atomicAdd(dp + 3, hv.w * v);
}

// ---------------------------------------------------------------------------
// Kernel 5: fused FC.  OUT = act( H@W1^T + b1 + sqrt(AGG)@W2^T + b2 )
// Same WMMA tiling as pool GEMM, two K loops into one accumulator.
// ---------------------------------------------------------------------------
__global__ void sage_fc_fused(const float* __restrict__ H, const float* __restrict__ AGG,
                              const float* __restrict__ W1, const float* __restrict__ B1,
                              const float* __restrict__ W2, const float* __restrict__ B2,
                              float* __restrict__ OUT,
                              int nrows, int K, int outc, int do_relu) {
    const int lane = threadIdx.x & 31;
    const int wave = threadIdx.x >> 5;
    const int colTiles = outc >> 4;
    const int rowTiles = nrows >> 4;
    int tile = blockIdx.x * (blockDim.x >> 5) + wave;
    if (tile >= rowTiles * colTiles) return;      // wave-uniform
    const int rt = tile / colTiles, ct = tile % colTiles;
    const int row0 = rt << 4, c0 = ct << 4;
    const int half = lane >> 4;
    const int lid  = lane & 15;

    const float* a1 = H   + (size_t)(row0 + lid) * K + 2 * half;
    const float* a2 = AGG + (size_t)(row0 + lid) * K + 2 * half;
    const float* b1 = W1  + (size_t)(c0 + lid) * K + 2 * half;
    const float* b2 = W2  + (size_t)(c0 + lid) * K + 2 * half;

    v8f acc = {};
    for (int k = 0; k < K; k += 4) {              // H @ W1^T
        v2f a, b;
        a.x = a1[k];     a.y = a1[k + 1];
        b.x = b1[k];     b.y = b1[k + 1];
        acc = __builtin_amdgcn_wmma_f32_16x16x4_f32(false, a, false, b,
                                                    (short)0, acc, false, false);
    }
    for (int k = 0; k < K; k += 4) {              // sqrt(AGG) @ W2^T  (MU root)
        v2f a, b;
        a.x = sqrtf(fmaxf(a2[k],     0.f));
        a.y = sqrtf(fmaxf(a2[k + 1], 0.f));
        b.x = b2[k];     b.y = b2[k + 1];
        acc = __builtin_amdgcn_wmma_f32_16x16x4_f32(false, a, false, b,
                                                    (short)0, acc, false, false);
    }

    const float bv = B1[c0 + lid] + B2[c0 + lid];
    float* op = OUT + (size_t)(row0 + 8 * half) * outc + c0 + lid;
    #pragma unroll
    for (int j = 0; j < 8; ++j) {
        float v = acc[j] + bv;
        if (do_relu) v = fmaxf(v, 0.f);
        op[(size_t)j * outc] = v;
    }
}

// ---------------------------------------------------------------------------
// Host-side orchestration
// ---------------------------------------------------------------------------
extern "C" void kernel_launch(void* const* d_in, const int* in_sizes, int n_in,
                              void* d_out, int out_size, void* d_ws, size_t ws_size,
                              hipStream_t stream) {
    const int IN_F = 128, HID = 128, OUT_F = 64;
    const float* x    = (const float*)d_in[0];
    const int*   esrc = (const int*)  d_in[1];
    const int*   edst = (const int*)  d_in[2];
    const float* eval = (const float*)d_in[3];
    const float* pw0  = (const float*)d_in[4];
    const float* pb0  = (const float*)d_in[5];
    const float* f1w0 = (const float*)d_in[6];
    const float* f1b0 = (const float*)d_in[7];
    const float* f2w0 = (const float*)d_in[8];
    const float* f2b0 = (const float*)d_in[9];
    const float* pw1  = (const float*)d_in[10];
    const float* pb1  = (const float*)d_in[11];
    const float* f1w1 = (const float*)d_in[12];
    const float* f1b1 = (const float*)d_in[13];
    const float* f2w1 = (const float*)d_in[14];
    const float* f2b1 = (const float*)d_in[15];

    const int N = in_sizes[0] / IN_F;
    const int E = in_sizes[1];
    float* outp = (float*)d_out;

    // workspace carve-up (256B aligned)
    size_t featBytes = ((size_t)N * 128 * sizeof(float) + 255) & ~(size_t)255;
    char* ws = (char*)d_ws;
    float* hbuf  = (float*)(ws);                       // pool output (both layers)
    float* hsq   = (float*)(ws + featBytes);           // squared pool output
    float* agg   = (float*)(ws + 2 * featBytes);       // scatter accumulator
    float* h1    = (float*)(ws + 3 * featBytes);       // layer-0 output
    float* invn  = (float*)(ws + 4 * featBytes);       // row inverse norms

    const int TPB = 256;                               // 8 waves/block
    const int wavesPerBlk = TPB >> 5;
    const int normBlocks  = (N + wavesPerBlk - 1) / wavesPerBlk;
    const int spmmBlocks  = (E + wavesPerBlk - 1) / wavesPerBlk;
    const size_t aggN = (size_t)N * 128;

    auto gemmBlocks = [&](int outc) {
        int tiles = (N >> 4) * (outc >> 4);
        return (tiles + wavesPerBlk - 1) / wavesPerBlk;
    };

    // ---------------- Layer 0 ----------------
    sage_rownorm<<<normBlocks, TPB, 0, stream>>>(x, invn, N, IN_F);
    sage_pool_gemm<<<gemmBlocks(IN_F), TPB, 0, stream>>>(x, invn, pw0, pb0,
                                                         hbuf, hsq, N, IN_F, IN_F);
    sage_zero<<<2048, TPB, 0, stream>>>(agg, aggN);
    sage_spmm<<<spmmBlocks, TPB, 0, stream>>>(hsq, esrc, edst, eval, agg, E, IN_F);
    sage_fc_fused<<<gemmBlocks(HID), TPB, 0, stream>>>(hbuf, agg, f1w0, f1b0,
                                                       f2w0, f2b0, h1,
                                                       N, IN_F, HID, /*relu=*/1);

    // ---------------- Layer 1 ----------------
    sage_rownorm<<<normBlocks, TPB, 0, stream>>>(h1, invn, N, HID);
    sage_pool_gemm<<<gemmBlocks(HID), TPB, 0, stream>>>(h1, invn, pw1, pb1,
                                                        hbuf, hsq, N, HID, HID);
    sage_zero<<<2048, TPB, 0, stream>>>(agg, aggN);
    sage_spmm<<<spmmBlocks, TPB, 0, stream>>>(hsq, esrc, edst, eval, agg, E, HID);
    sage_fc_fused<<<gemmBlocks(OUT_F), TPB, 0, stream>>>(hbuf, agg, f1w1, f1b1,
                                                         f2w1, f2b1, outp,
                                                         N, HID, OUT_F, /*relu=*/0);
}